// MlsDeformModule_37503654428936
// MI455X (gfx1250) — compile-verified
//
#include <hip/hip_runtime.h>
#include <math.h>

// MLS rigid deformation -> dense remap grid (H, W, 2), H = W = 768, N = 64.
//
// Strategy (see analysis): reduce per-pixel work to 9 weighted moments and
// compute them as a GEMM  wgt(16px x 64) @ Bmom(64 x 9)  using chained
// V_WMMA_F32_16X16X4_F32 (the only f32 WMMA on CDNA5; f16/fp8 would overflow
// on d^2 ~ 768^2 and w ~ 1e9). Weights are computed on VALU with v_rcp_f32.

typedef __attribute__((ext_vector_type(2))) float v2f;
typedef __attribute__((ext_vector_type(8))) float v8f;

#define WIMG 768
#define HIMG 768
#define SEGS 12           // 768 / 64 pixels per wave
#define TILES_PER_WAVE 4  // 4 x 16-pixel WMMA tiles
#define WAVES_PER_BLOCK 8

// moment component j of control point (px,py)->(qx,qy):
// [1, px, py, qx, qy, qx*px, qx*py, qy*px, qy*py, 0...]
__device__ __forceinline__ float moment_component(int j, float px, float py,
                                                  float qx, float qy) {
  switch (j) {
    case 0: return 1.0f;
    case 1: return px;
    case 2: return py;
    case 3: return qx;
    case 4: return qy;
    case 5: return qx * px;
    case 6: return qx * py;
    case 7: return qy * px;
    case 8: return qy * py;
    default: return 0.0f;  // pad columns 9..15 of the WMMA tile with zeros
  }
}

// Build per-lane tables in exact WMMA B-matrix / A-feed order.
// Index = t*32 + lane, lane = (j = N/M index, h = K-half).
// B tile t, VGPR0/1 for lane(j,h) = moment_j(ctrl 4t+2h), moment_j(ctrl 4t+2h+1)
// P tile t for lane(j,h)          = (pi[4t+2h].xy, pi[4t+2h+1].xy)
__global__ void build_tables(const float* __restrict__ pi,
                             const float* __restrict__ qi,
                             float2* __restrict__ Btab,
                             float4* __restrict__ Ptab) {
  int idx = threadIdx.x;  // 0..511
  int lane = idx & 31;
  int t = idx >> 5;
  int j = lane & 15;
  int h = lane >> 4;
  int c0 = 4 * t + 2 * h;
  int c1 = c0 + 1;
  float px0 = pi[2 * c0 + 0], py0 = pi[2 * c0 + 1];
  float qx0 = qi[2 * c0 + 0], qy0 = qi[2 * c0 + 1];
  float px1 = pi[2 * c1 + 0], py1 = pi[2 * c1 + 1];
  float qx1 = qi[2 * c1 + 0], qy1 = qi[2 * c1 + 1];
  Btab[idx] = make_float2(moment_component(j, px0, py0, qx0, qy0),
                          moment_component(j, px1, py1, qx1, qy1));
  Ptab[idx] = make_float4(px0, py0, px1, py1);
}

__global__ __launch_bounds__(256) void mls_wmma_kernel(
    const float2* __restrict__ Btab, const float4* __restrict__ Ptab,
    float* __restrict__ out) {
  __shared__ float xch[WAVES_PER_BLOCK][16 * 16];  // per-wave transpose buffer

  const int lane = threadIdx.x & 31;
  const int wave = threadIdx.x >> 5;
  const int gwave = blockIdx.x * WAVES_PER_BLOCK + wave;  // 0..9215
  const int y = gwave / SEGS;
  const int xbase = (gwave % SEGS) * 64;
  const int j = lane & 15;
  const int h = lane >> 4;
  const float ys = (float)y;

  // Wave-constant WMMA B operand (moment matrix) + control-point pairs,
  // preloaded once and reused across 4 pixel tiles.
  v2f b[16];
  float4 p[16];
#pragma unroll
  for (int t = 0; t < 16; ++t) {
    float2 bb = Btab[t * 32 + lane];
    b[t][0] = bb.x;
    b[t][1] = bb.y;
    p[t] = Ptab[t * 32 + lane];
  }

  float* xb = &xch[wave][0];

  for (int tile = 0; tile < TILES_PER_WAVE; ++tile) {
    const float xs = (float)(xbase + tile * 16 + j);  // pixel x for A row j
    v8f acc = {0.f, 0.f, 0.f, 0.f, 0.f, 0.f, 0.f, 0.f};

#pragma unroll
    for (int t = 0; t < 16; ++t) {
      // weights w = 1/(|pi - v|^2 + 1e-9) in A-matrix layout (K-slice of 4)
      float dx0 = p[t].x - xs, dy0 = p[t].y - ys;
      float dx1 = p[t].z - xs, dy1 = p[t].w - ys;
      v2f a;
      a[0] = __builtin_amdgcn_rcpf(fmaf(dx0, dx0, fmaf(dy0, dy0, 1e-9f)));
      a[1] = __builtin_amdgcn_rcpf(fmaf(dx1, dx1, fmaf(dy1, dy1, 1e-9f)));
      // D(16px x 16comp) += A(16x4 weights) * B(4x16 moments)
      acc = __builtin_amdgcn_wmma_f32_16x16x4_f32(
          false, a, false, b[t], (short)0, acc, false, false);
    }

    // D layout: lane(j,h) VGPR r = component j of pixel (r + 8h).
    // Transpose through LDS so each lane gets all 9 components of one pixel.
#pragma unroll
    for (int r = 0; r < 8; ++r) xb[(r + 8 * h) * 16 + j] = acc[r];
    asm volatile("s_wait_dscnt 0" ::: "memory");  // same-wave DS is in-order

    if (lane < 16) {
      const float4 mA = *(const float4*)&xb[lane * 16 + 0];  // S,Spx,Spy,Sqx
      const float4 mB = *(const float4*)&xb[lane * 16 + 4];  // Sqy,Mxx,Mxy,Myx
      const float Myy = xb[lane * 16 + 8];
      const float S = mA.x, Spx = mA.y, Spy = mA.z, Sqx = mA.w;
      const float Sqy = mB.x, Mxx = mB.y, Mxy = mB.z, Myx = mB.w;

      const float rs = __builtin_amdgcn_rcpf(S);
      const float psx = Spx * rs, psy = Spy * rs;
      const float qsx = Sqx * rs, qsy = Sqy * rs;

      const float xpix = (float)(xbase + tile * 16 + lane);
      const float vpx = xpix - psx;
      const float vpy = ys - psy;

      const float alpha = (Mxx + Myy) - (Sqx * Spx + Sqy * Spy) * rs;
      const float beta = (Mxy - Myx) - (Sqx * Spy - Sqy * Spx) * rs;

      const float frvx = alpha * vpx + beta * vpy;
      const float frvy = alpha * vpy - beta * vpx;

      const float vpn = sqrtf(vpx * vpx + vpy * vpy);
      const float frvn = sqrtf(frvx * frvx + frvy * frvy) + 1e-10f;
      const float s = vpn / frvn;

      // out[y][x] = fv ; contiguous float2 row stores (coalesced 128B)
      ((float2*)out)[y * WIMG + xbase + tile * 16 + lane] =
          make_float2(s * frvx + qsx, s * frvy + qsy);
    }
    asm volatile("s_wait_dscnt 0" ::: "memory");
  }
}

extern "C" void kernel_launch(void* const* d_in, const int* in_sizes, int n_in,
                              void* d_out, int out_size, void* d_ws,
                              size_t ws_size, hipStream_t stream) {
  (void)in_sizes; (void)n_in; (void)out_size; (void)ws_size;
  // setup_inputs order: img (unused -> only shape), pi (64x2 f32), qi (64x2 f32)
  const float* pi = (const float*)d_in[1];
  const float* qi = (const float*)d_in[2];
  float* out = (float*)d_out;

  float2* Btab = (float2*)d_ws;                                   // 512 * 8B
  float4* Ptab = (float4*)((char*)d_ws + 512 * sizeof(float2));   // 512 * 16B

  build_tables<<<1, 512, 0, stream>>>(pi, qi, Btab, Ptab);

  const int nwaves = HIMG * SEGS;                 // 9216 waves
  const int nblocks = nwaves / WAVES_PER_BLOCK;   // 1152 blocks of 8 waves
  mls_wmma_kernel<<<nblocks, 256, 0, stream>>>(Btab, Ptab, out);
}